// AClayer_69569880261140
// MI455X (gfx1250) — compile-verified
//
#include <hip/hip_runtime.h>
#include <hip/hip_bf16.h>

// Problem constants (from reference): n=4, c=256, h=w=128
#define NB   4
#define CH   256
#define HH   128
#define WW   128
#define HW   (HH * WW)          // 16384
#define PLANES (NB * CH)        // 1024
#define POOL 256                // 16*16 pooled positions
#define OC   9                  // conv1 out channels (Ck*K*K)
#define EPSV 1e-5f

typedef __attribute__((ext_vector_type(2))) float v2f;
typedef __attribute__((ext_vector_type(4))) float v4f;
typedef __attribute__((ext_vector_type(8))) float v8f;

// ---------------------------------------------------------------------------
// Kernel 1: AdaptiveAvgPool2d((16,16)) with 8x8 windows.
// One block per (n,c) plane; thread t owns pooled cell (t/16, t%16).
// Reads are float4-coalesced; this pass also warms x into the 192 MB L2.
// Output is stored TRANSPOSED as xpT[n][p][c] so the GEMM's B-fragment
// K-pair (kb, kb+1) is one contiguous b64 load.
// ---------------------------------------------------------------------------
__global__ void avgpool_k(const float* __restrict__ x, float* __restrict__ xpT) {
    int plane = blockIdx.x;                       // n*256 + c
    int n = plane >> 8;
    int c = plane & 255;
    const float* xin = x + (size_t)plane * HW;
    int tid = threadIdx.x;                        // 0..255  (= pooled pos p)
    int ty = tid >> 4, tx = tid & 15;
    const float* base = xin + (ty * 8) * WW + tx * 8;
    float sum = 0.f;
#pragma unroll
    for (int r = 0; r < 8; ++r) {
        v4f a = *(const v4f*)(base + r * WW);
        v4f b = *(const v4f*)(base + r * WW + 4);
        sum += a.x + a.y + a.z + a.w + b.x + b.y + b.z + b.w;
    }
    xpT[(size_t)n * (POOL * CH) + (size_t)tid * CH + c] = sum * (1.0f / 64.0f);
}

// ---------------------------------------------------------------------------
// Kernel 2: sigma[n, 9, 256] = w_conv(9x256) @ xp[n](256ch x 256pos)
// V_WMMA_F32_16X16X4_F32 (exact f32 semantics): M=16 (9 zero-padded),
// N=16 tile, K stepped by 4. One wave per (n, ntile); 4*16 = 64 blocks.
// w_conv is staged into LDS padded to 16 rows -> the hot loop is branchless:
// ds_load_b64 (A pair) + global_load_b64 (B pair) + v_wmma per step.
// A layout: lanes l / l+16 hold M=l; vgpr0/1 = K+0/K+1 (upper half K+2/K+3).
// B mirrors A with N across lanes. D: 8 VGPRs, M = r + 8*half.
// ---------------------------------------------------------------------------
__global__ void gemm_sigma_k(const float* __restrict__ w,
                             const float* __restrict__ xpT,
                             float* __restrict__ sigma) {
    __shared__ float aw[16 * CH];     // 16 KB: w_conv padded to 16x256
    int bid  = blockIdx.x;            // 0..63
    int n    = bid >> 4;
    int pt   = bid & 15;
    int lane = threadIdx.x;           // 0..31 (one wave)

    // Cooperative padded fill: rows 0..8 from w, rows 9..15 zero.
    for (int i = lane * 4; i < 16 * CH; i += 32 * 4) {
        int row = i >> 8;
        v4f v;
        if (row < OC) v = *(const v4f*)(w + i);
        else          { v.x = 0.f; v.y = 0.f; v.z = 0.f; v.w = 0.f; }
        *(v4f*)(aw + i) = v;
    }
    __syncthreads();

    int half = lane >> 4;
    int l    = lane & 15;
    int p    = pt * 16 + l;           // pooled position (N index)
    const float* xb = xpT + (size_t)n * (POOL * CH) + (size_t)p * CH;

    v8f acc = {};
#pragma unroll 4
    for (int k = 0; k < CH; k += 4) {
        int kb = k + 2 * half;
        v2f a = *(const v2f*)(aw + l * CH + kb);   // A[M=l, kb..kb+1] (LDS)
        v2f b = *(const v2f*)(xb + kb);            // B[kb..kb+1, N=p]
        acc = __builtin_amdgcn_wmma_f32_16x16x4_f32(
            /*neg_a=*/false, a, /*neg_b=*/false, b,
            /*c_mod=*/(short)0, acc, /*reuse_a=*/false, /*reuse_b=*/false);
    }
#pragma unroll
    for (int r = 0; r < 8; ++r) {
        int m = r + 8 * half;                      // D-matrix row
        if (m < OC) sigma[((size_t)n * OC + m) * POOL + p] = acc[r];
    }
}

// ---------------------------------------------------------------------------
// Kernel 3: BatchNorm2d (training stats, biased var) over (n, 16, 16) per
// channel m of 9, then write normalized snorm[n, m, p].
// 9 blocks x 256 threads; thread t = pooled position p, covers all 4 batches.
// ---------------------------------------------------------------------------
__global__ void bn_k(const float* __restrict__ sigma,
                     const float* __restrict__ gamma,
                     const float* __restrict__ beta,
                     float* __restrict__ snorm) {
    __shared__ float ssum[256];
    __shared__ float ssq[256];
    int m = blockIdx.x;               // 0..8
    int t = threadIdx.x;              // 0..255  (= p)
    float v[NB];
    float s = 0.f, sq = 0.f;
#pragma unroll
    for (int n = 0; n < NB; ++n) {
        float val = sigma[((size_t)n * OC + m) * POOL + t];
        v[n] = val; s += val; sq += val * val;
    }
    ssum[t] = s; ssq[t] = sq;
    __syncthreads();
    for (int off = 128; off > 0; off >>= 1) {
        if (t < off) { ssum[t] += ssum[t + off]; ssq[t] += ssq[t + off]; }
        __syncthreads();
    }
    const float inv_n = 1.0f / (NB * POOL);
    float mean  = ssum[0] * inv_n;
    float var   = ssq[0] * inv_n - mean * mean;
    float scale = gamma[m] * rsqrtf(var + EPSV);
    float shift = beta[m] - mean * scale;
#pragma unroll
    for (int n = 0; n < NB; ++n)
        snorm[((size_t)n * OC + m) * POOL + t] = v[n] * scale + shift;
}

// ---------------------------------------------------------------------------
// Kernel 4: data-dependent 3x3 depthwise conv (zero padding).
// Channel index c selects the pooled-position row of snorm (the broadcast
// trick in the reference). One block per (n,c) plane; each thread computes
// 4 consecutive pixels from a 3x6 register window (aligned float4 centers).
// x re-reads hit L2 (64 MiB resident); `out` uses NON-TEMPORAL stores so the
// 64 MiB of write traffic does not evict x from L2 mid-pass.
// ---------------------------------------------------------------------------
__global__ void dwconv_k(const float* __restrict__ x,
                         const float* __restrict__ snorm,
                         float* __restrict__ out) {
    int plane = blockIdx.x;                // n*256 + c
    int n = plane >> 8;
    int c = plane & 255;
    const float* xin = x + (size_t)plane * HW;
    float* o = out + (size_t)plane * HW;

    float s[9];
#pragma unroll
    for (int k = 0; k < 9; ++k)            // uniform per block -> scalar loads
        s[k] = snorm[((size_t)n * OC + k) * POOL + c];

    int tid = threadIdx.x;                 // 256 threads
    // 32 quads per row * 128 rows = 4096 quads; 16 iterations of 256 threads.
#pragma unroll 2
    for (int it = 0; it < 16; ++it) {
        int q  = it * 256 + tid;
        int y  = q >> 5;                   // 0..127
        int x0 = (q & 31) << 2;            // 0,4,...,124

        float row[3][6];
#pragma unroll
        for (int dy = 0; dy < 3; ++dy) {
            int yy = y + dy - 1;
            if ((unsigned)yy < (unsigned)HH) {
                const float* rp = xin + yy * WW + x0;
                v4f mid = *(const v4f*)rp;                // 16B-aligned
                row[dy][1] = mid.x; row[dy][2] = mid.y;
                row[dy][3] = mid.z; row[dy][4] = mid.w;
                row[dy][0] = (x0 > 0)      ? rp[-1] : 0.f;
                row[dy][5] = (x0 < WW - 4) ? rp[4]  : 0.f;
            } else {
#pragma unroll
                for (int dx = 0; dx < 6; ++dx) row[dy][dx] = 0.f;
            }
        }

        v4f r;
#pragma unroll
        for (int px = 0; px < 4; ++px) {
            float acc = 0.f;
#pragma unroll
            for (int i = 0; i < 3; ++i)
#pragma unroll
                for (int j = 0; j < 3; ++j)
                    acc = fmaf(s[i * 3 + j], row[i][px + j], acc);
            r[px] = acc;
        }
        __builtin_nontemporal_store(r, (v4f*)(o + y * WW + x0));
    }
}

// ---------------------------------------------------------------------------
extern "C" void kernel_launch(void* const* d_in, const int* in_sizes, int n_in,
                              void* d_out, int out_size, void* d_ws, size_t ws_size,
                              hipStream_t stream) {
    const float* x     = (const float*)d_in[0];   // (4,256,128,128)
    const float* wconv = (const float*)d_in[1];   // (9,256)
    const float* gamma = (const float*)d_in[2];   // (9,)
    const float* beta  = (const float*)d_in[3];   // (9,)
    float* out = (float*)d_out;                   // (4,256,128,128)

    float* ws    = (float*)d_ws;
    float* xpT   = ws;                            // 4*256*256   = 262144 floats
    float* sigma = xpT + (size_t)NB * POOL * CH;  // 4*9*256     = 9216 floats
    float* snorm = sigma + (size_t)NB * OC * POOL;

    avgpool_k   <<<PLANES, 256, 0, stream>>>(x, xpT);
    gemm_sigma_k<<<NB * 16, 32,  0, stream>>>(wconv, xpT, sigma);
    bn_k        <<<OC,      256, 0, stream>>>(sigma, gamma, beta, snorm);
    dwconv_k    <<<PLANES,  256, 0, stream>>>(x, snorm, out);
}